// MinEntropyConsensusLoss_71004399337949
// MI455X (gfx1250) — compile-verified
//
#include <hip/hip_runtime.h>
#include <math.h>

// ---- CDNA5 (gfx1250, wave32) types ---------------------------------------
typedef __attribute__((ext_vector_type(2))) float v2f;
typedef __attribute__((ext_vector_type(8))) float v8f;

#define COLS 256
#define TPB 256          // 8 wave32 waves per block
#define WAVES_PER_BLK (TPB / 32)
#define NBLK 2048        // 16384 waves -> 8 rows/wave at B=131072; 8KB of d_ws

__device__ __forceinline__ float wave_max32(float v) {
#pragma unroll
  for (int m = 1; m < 32; m <<= 1)
    v = fmaxf(v, __shfl_xor(v, m, 32));
  return v;
}

// One wave32 per row. lse_x + lse_y - max(x+y), accumulated per block.
__global__ __launch_bounds__(TPB) void mecl_rows(const float* __restrict__ x,
                                                 const float* __restrict__ y,
                                                 float* __restrict__ partial,
                                                 int rows) {
  __shared__ float s_wave[WAVES_PER_BLK];
  const int lane  = threadIdx.x & 31;
  const int wave  = threadIdx.x >> 5;
  const int gwave = blockIdx.x * WAVES_PER_BLK + wave;
  const int nwave = gridDim.x * WAVES_PER_BLK;

  // Constant 0/1 B-matrix (4x16) for the packed (Sx,Sy) reduction WMMA:
  // B[k even, n<8] = 1, B[k odd, n>=8] = 1, else 0.
  v2f bsel;
  {
    const int n = lane & 15;
    bsel.x = (n < 8) ? 1.0f : 0.0f;  // k = 0 (lanes 0-15) / k = 2 (lanes 16-31)
    bsel.y = 1.0f - bsel.x;          // k = 1 / k = 3
  }

  float acc = 0.0f;

  for (int row = gwave; row < rows; row += nwave) {
    const float4* xr = (const float4*)(x + (size_t)row * COLS);
    const float4* yr = (const float4*)(y + (size_t)row * COLS);
    // Fully coalesced: lanes cover 512B per load instruction.
    float4 x0 = xr[lane], x1 = xr[lane + 32];
    float4 y0 = yr[lane], y1 = yr[lane + 32];

    if (row + nwave < rows) {  // global_prefetch_b8 for the next row
      __builtin_prefetch(x + (size_t)(row + nwave) * COLS + lane * 8, 0, 0);
      __builtin_prefetch(y + (size_t)(row + nwave) * COLS + lane * 8, 0, 0);
    }

    float xa[8] = {x0.x, x0.y, x0.z, x0.w, x1.x, x1.y, x1.z, x1.w};
    float ya[8] = {y0.x, y0.y, y0.z, y0.w, y1.x, y1.y, y1.z, y1.w};

    // Row maxes (softmax stabilization) and max(x+y) via lane butterflies.
    float mx = -INFINITY, my = -INFINITY, mt = -INFINITY;
#pragma unroll
    for (int j = 0; j < 8; ++j) {
      mx = fmaxf(mx, xa[j]);
      my = fmaxf(my, ya[j]);
      mt = fmaxf(mt, xa[j] + ya[j]);
    }
    mx = wave_max32(mx);
    my = wave_max32(my);
    mt = wave_max32(mt);

    // Per-lane partial sums of exp.
    float sx = 0.0f, sy = 0.0f;
#pragma unroll
    for (int j = 0; j < 8; ++j) {
      sx += __expf(xa[j] - mx);
      sy += __expf(ya[j] - my);
    }

    // Exact-f32 cross-lane sums of both sx and sy with ONE
    // v_wmma_f32_16x16x4_f32:
    //   A[m,0]=sx(lane m)  A[m,1]=sy(lane m)
    //   A[m,2]=sx(lane m+16) A[m,3]=sy(lane m+16)   (native A layout)
    //   D[m,n<8] = sx[m]+sx[m+16],  D[m,n>=8] = sy[m]+sy[m+16]
    v2f a;
    a.x = sx;
    a.y = sy;
    v8f c = {};  // zero accumulator
    v8f d = __builtin_amdgcn_wmma_f32_16x16x4_f32(
        /*neg_a=*/false, a, /*neg_b=*/false, bsel,
        /*c_mod=*/(short)0, c, /*reuse_a=*/false, /*reuse_b=*/false);

    // Sum this lane's column of D (rows 0-7 or 8-15), then fold halves:
    float colsum = d[0] + d[1] + d[2] + d[3] + d[4] + d[5] + d[6] + d[7];
    float v = colsum + __shfl_xor(colsum, 16, 32);
    // lanes 0-7 now hold Sx (full wave sum), lanes 8-15 hold Sy.
    float vSy = __shfl_xor(v, 8, 32);  // lane 0 reads lane 8 -> Sy

    if (lane == 0) {
      // row value: lse_x + lse_y - max(x+y)
      acc += mx + my + __logf(v) + __logf(vSy) - mt;
    }
  }

  if (lane == 0) s_wave[wave] = acc;
  __syncthreads();
  if (threadIdx.x == 0) {
    float t = 0.0f;
#pragma unroll
    for (int w = 0; w < WAVES_PER_BLK; ++w) t += s_wave[w];
    partial[blockIdx.x] = t;
  }
}

// Deterministic fixed-order final reduction of the per-block partials.
__global__ __launch_bounds__(256) void mecl_reduce(const float* __restrict__ partial,
                                                   float* __restrict__ out,
                                                   int nblk, float scale) {
  __shared__ float s[256];
  float v = 0.0f;
  for (int i = threadIdx.x; i < nblk; i += 256) v += partial[i];
  s[threadIdx.x] = v;
  __syncthreads();
  for (int off = 128; off > 0; off >>= 1) {
    if ((int)threadIdx.x < off) s[threadIdx.x] += s[threadIdx.x + off];
    __syncthreads();
  }
  if (threadIdx.x == 0) out[0] = s[0] * scale;
}

extern "C" void kernel_launch(void* const* d_in, const int* in_sizes, int n_in,
                              void* d_out, int out_size, void* d_ws, size_t ws_size,
                              hipStream_t stream) {
  const float* x = (const float*)d_in[0];
  const float* y = (const float*)d_in[1];
  float* out = (float*)d_out;
  float* partial = (float*)d_ws;  // NBLK floats = 8 KB

  const int rows = in_sizes[0] / COLS;  // 131072 for the reference shapes

  mecl_rows<<<NBLK, TPB, 0, stream>>>(x, y, partial, rows);
  mecl_reduce<<<1, 256, 0, stream>>>(partial, out, NBLK, 0.5f / (float)rows);
}